// LinearAttention_39135742001367
// MI455X (gfx1250) — compile-verified
//
#include <hip/hip_runtime.h>
#include <cstddef>

typedef __attribute__((ext_vector_type(2))) float v2f;
typedef __attribute__((ext_vector_type(8))) float v8f;
typedef __attribute__((ext_vector_type(4))) int   v4i_t;
typedef v4i_t __attribute__((address_space(1))) g_v4i;   // global-AS 16B vector
typedef v4i_t __attribute__((address_space(3))) l_v4i;   // LDS-AS 16B vector

#define NPIX   16384   // H*W
#define CDIM   128
#define NBATCH 16
#define NCHUNK 1024    // split-K chunk along n for the S GEMM
#define NCHUNKS (NPIX / NCHUNK)  // 16
#define NSTAGES (NCHUNK / 32)    // 32
#define LDSPAD 36                // row stride (floats): keeps float4 16B-aligned,
                                 // 36*l16 mod 64 distinct -> conflict-free WMMA reads

#if defined(__has_builtin)
#if __has_builtin(__builtin_amdgcn_global_load_async_to_lds_b128)
#define HAVE_ASYNC_LDS 1
#endif
#endif

__device__ __forceinline__ v8f wmma_f32(v2f a, v2f b, v8f c) {
    return __builtin_amdgcn_wmma_f32_16x16x4_f32(false, a, false, b, (short)0, c,
                                                 false, false);
}

__device__ __forceinline__ void wait_async0() {
#if defined(HAVE_ASYNC_LDS)
#if __has_builtin(__builtin_amdgcn_s_wait_asynccnt)
    __builtin_amdgcn_s_wait_asynccnt(0);
#else
    asm volatile("s_wait_asynccnt 0" ::: "memory");
#endif
#endif
}

// 16-byte global -> LDS copy: async DMA path on CDNA5, plain store fallback.
__device__ __forceinline__ void copy16_g2l(const float* g, float* l) {
#if defined(HAVE_ASYNC_LDS)
    __builtin_amdgcn_global_load_async_to_lds_b128(
        (g_v4i*)g, (l_v4i*)l, 0, 0);
#else
    *(float4*)l = *(const float4*)g;
#endif
}

// ---------------------------------------------------------------------------
// GEMM: out[b][m][n] = sum_c A[m][c] * x[b][c][n]  (+ bias[m] if BIAS)
// A: 128x128 row-major (per-batch stride aStride). x: (b, 128, 16384).
// Block = 256 threads = 8 waves; each wave owns one 16-row strip and the
// block's 32-wide n tile (two 16x16 WMMA tiles), K-stepped by 4.
// ---------------------------------------------------------------------------
template <bool BIAS>
__global__ void gemm_wmma_kernel(const float* __restrict__ Abase, size_t aStride,
                                 const float* __restrict__ x,
                                 const float* __restrict__ bias,
                                 float* __restrict__ out) {
    const int b    = blockIdx.y;
    const int n0   = blockIdx.x * 32;
    const int tid  = threadIdx.x;
    const int lane = tid & 31;
    const int wv   = tid >> 5;       // 0..7
    const int l16  = lane & 15;
    const int half = lane >> 4;      // 0 or 1
    const int m0   = wv * 16;

    const float* A  = Abase + (size_t)b * aStride;
    const float* xb = x + (size_t)b * CDIM * NPIX;
    const float* arow = A + (size_t)(m0 + l16) * CDIM;

    // One prefetch instruction touches the whole 128x32 x panel for this block
    // (per-lane addresses: 2 lanes per c-row, 64B apart) -> global_prefetch_b8.
    {
        const int prow = tid >> 1;
        const int pseg = (tid & 1) * 16;
        __builtin_prefetch(xb + (size_t)prow * NPIX + n0 + pseg, 0, 3);
    }

    v8f acc0 = {}, acc1 = {};
#pragma unroll 4
    for (int kk = 0; kk < CDIM; kk += 4) {
        const int kr = kk + 2 * half;
        v2f a;
        a.x = arow[kr];
        a.y = arow[kr + 1];
        const float* xr0 = xb + (size_t)kr * NPIX + n0;
        const float* xr1 = xr0 + NPIX;
        v2f b0, b1;
        b0.x = xr0[l16];       b0.y = xr1[l16];
        b1.x = xr0[16 + l16];  b1.y = xr1[16 + l16];
        acc0 = wmma_f32(a, b0, acc0);
        acc1 = wmma_f32(a, b1, acc1);
    }

    float* ob = out + (size_t)b * CDIM * NPIX;
#pragma unroll
    for (int r = 0; r < 8; ++r) {
        const int row = m0 + r + 8 * half;
        const float bv = BIAS ? bias[row] : 0.0f;
        ob[(size_t)row * NPIX + n0 + l16]      = acc0[r] + bv;
        ob[(size_t)row * NPIX + n0 + 16 + l16] = acc1[r] + bv;
    }
}

// ---------------------------------------------------------------------------
// Per-row softmax stats over n=16384: row max and 1/sum(exp(k-max)).
// One 256-thread block per (b,d) row. k should be L2-resident (134MB < 192MB).
// ---------------------------------------------------------------------------
__global__ void softmax_stats_kernel(const float* __restrict__ kbuf,
                                     float* __restrict__ rmax,
                                     float* __restrict__ rinv) {
    __shared__ float red[256];
    const int row = blockIdx.x;           // b*128 + d
    const int tid = threadIdx.x;
    const float* p = kbuf + (size_t)row * NPIX;

    float m = -3.402823466e38f;
    for (int i = tid; i < NPIX; i += 256) m = fmaxf(m, p[i]);
    red[tid] = m;
    __syncthreads();
    for (int s = 128; s > 0; s >>= 1) {
        if (tid < s) red[tid] = fmaxf(red[tid], red[tid + s]);
        __syncthreads();
    }
    m = red[0];
    __syncthreads();

    float sum = 0.0f;
    for (int i = tid; i < NPIX; i += 256) sum += __expf(p[i] - m);
    red[tid] = sum;
    __syncthreads();
    for (int s = 128; s > 0; s >>= 1) {
        if (tid < s) red[tid] += red[tid + s];
        __syncthreads();
    }
    if (tid == 0) {
        rmax[row] = m;
        rinv[row] = 1.0f / red[0];
    }
}

// ---------------------------------------------------------------------------
// Split-K partial: Spart[chunk][b][d][c] = sum_{n in chunk} ksoft[b][d][n]*x[b][c][n]
// 64(d) x 64(c) block tile, 8 waves * 2 WMMA tiles. Double-buffered LDS:
//  - x panel for stage i+1 moved by GLOBAL_LOAD_ASYNC_TO_LDS_B128 (ASYNCcnt)
//  - k panel for stage i+1: global b128 loads before the WMMA block, exp +
//    ds_store_b128 after it, so both waits land behind the matrix math.
// ---------------------------------------------------------------------------
__global__ void s_partial_kernel(const float* __restrict__ kbuf,
                                 const float* __restrict__ x,
                                 const float* __restrict__ rmax,
                                 const float* __restrict__ rinv,
                                 float* __restrict__ Spart) {
    __shared__ float kt[2][64][LDSPAD];
    __shared__ float xt[2][64][LDSPAD];
    __shared__ float rm[64];
    __shared__ float ri[64];

    const int tid  = threadIdx.x;
    const int lane = tid & 31;
    const int wv   = tid >> 5;
    const int l16  = lane & 15;
    const int half = lane >> 4;

    const int b     = blockIdx.z;
    const int d0    = (blockIdx.x >> 1) * 64;
    const int c0    = (blockIdx.x & 1) * 64;
    const int chunk = blockIdx.y;
    const int nbase = chunk * NCHUNK;

    if (tid < 64) {
        const int dg = b * CDIM + d0 + tid;
        rm[tid] = rmax[dg];
        ri[tid] = rinv[dg];
    }
    __syncthreads();

    const int tm  = wv >> 1;          // 0..3  -> d strip
    const int tn0 = (wv & 1) * 32;    // 0/32  -> c strips (two tiles)

    const float* kb = kbuf + ((size_t)b * CDIM + d0) * NPIX;
    const float* xb = x    + ((size_t)b * CDIM + c0) * NPIX;

    // Per-thread staging assignment: 512 float4s per 64x32 panel, 2 per thread.
    int srow[2], scol[2];
#pragma unroll
    for (int j = 0; j < 2; ++j) {
        const int q = tid + j * 256;
        srow[j] = q >> 3;           // 0..63
        scol[j] = (q & 7) * 4;      // 0..28
    }

    v8f acc0 = {}, acc1 = {};
    float4 kv[2];

    // Prologue: stage 0 into buffer 0.
#pragma unroll
    for (int j = 0; j < 2; ++j) {
        copy16_g2l(xb + (size_t)srow[j] * NPIX + nbase + scol[j],
                   &xt[0][srow[j]][scol[j]]);
        kv[j] = *(const float4*)(kb + (size_t)srow[j] * NPIX + nbase + scol[j]);
    }
#pragma unroll
    for (int j = 0; j < 2; ++j) {
        const float m = rm[srow[j]], s = ri[srow[j]];
        float4 t;
        t.x = __expf(kv[j].x - m) * s;
        t.y = __expf(kv[j].y - m) * s;
        t.z = __expf(kv[j].z - m) * s;
        t.w = __expf(kv[j].w - m) * s;
        *(float4*)&kt[0][srow[j]][scol[j]] = t;
    }

    int buf = 0;
    for (int st = 0; st < NSTAGES; ++st) {
        wait_async0();          // my async x-panel for stage st landed
        __syncthreads();        // + everyone's k-panel stores; frees buf^1

        const bool has_next = (st + 1) < NSTAGES;
        if (has_next) {
            const int nsn = nbase + (st + 1) * 32;
#pragma unroll
            for (int j = 0; j < 2; ++j) {
                copy16_g2l(xb + (size_t)srow[j] * NPIX + nsn + scol[j],
                           &xt[buf ^ 1][srow[j]][scol[j]]);
                kv[j] = *(const float4*)(kb + (size_t)srow[j] * NPIX + nsn + scol[j]);
            }
        }

        const float (*ktb)[LDSPAD] = kt[buf];
        const float (*xtb)[LDSPAD] = xt[buf];
#pragma unroll
        for (int kk = 0; kk < 32; kk += 4) {
            const int kr = kk + 2 * half;
            v2f a, b0, b1;
            a.x  = ktb[tm * 16 + l16][kr];
            a.y  = ktb[tm * 16 + l16][kr + 1];
            b0.x = xtb[tn0 + l16][kr];
            b0.y = xtb[tn0 + l16][kr + 1];
            b1.x = xtb[tn0 + 16 + l16][kr];
            b1.y = xtb[tn0 + 16 + l16][kr + 1];
            acc0 = wmma_f32(a, b0, acc0);
            acc1 = wmma_f32(a, b1, acc1);
        }

        if (has_next) {
#pragma unroll
            for (int j = 0; j < 2; ++j) {
                const float m = rm[srow[j]], s = ri[srow[j]];
                float4 t;
                t.x = __expf(kv[j].x - m) * s;
                t.y = __expf(kv[j].y - m) * s;
                t.z = __expf(kv[j].z - m) * s;
                t.w = __expf(kv[j].w - m) * s;
                *(float4*)&kt[buf ^ 1][srow[j]][scol[j]] = t;
            }
        }
        buf ^= 1;
    }

    float* Sp = Spart + ((size_t)chunk * NBATCH + b) * (CDIM * CDIM);
#pragma unroll
    for (int r = 0; r < 8; ++r) {
        const int d = d0 + tm * 16 + r + 8 * half;
        Sp[d * CDIM + c0 + tn0 + l16]      = acc0[r];
        Sp[d * CDIM + c0 + tn0 + 16 + l16] = acc1[r];
    }
}

// Ssum[b][d][c] = sum over chunks of Spart
__global__ void s_reduce_kernel(const float* __restrict__ Spart,
                                float* __restrict__ Ssum) {
    const int idx = blockIdx.x * 256 + threadIdx.x;   // over 16*16384
    const int b   = idx >> 14;
    const int dc  = idx & 16383;
    float acc = 0.0f;
#pragma unroll
    for (int ch = 0; ch < NCHUNKS; ++ch)
        acc += Spart[((size_t)ch * NBATCH + b) * (CDIM * CDIM) + dc];
    Ssum[idx] = acc;
}

// ctx[b][h][d][e] = sum_c Ssum[b][h*32+d][c] * Wv[h*32+e][c]   (Wv = w_qkv rows 256..383)
__global__ void ctx_kernel(const float* __restrict__ Ssum,
                           const float* __restrict__ w_qkv,
                           float* __restrict__ ctx) {
    const int idx = blockIdx.x * 256 + threadIdx.x;   // over 16*4096
    const int b   = idx >> 12;
    const int h   = (idx >> 10) & 3;
    const int d   = (idx >> 5) & 31;
    const int e   = idx & 31;
    const float* srow = Ssum + (size_t)b * (CDIM * CDIM) + (h * 32 + d) * CDIM;
    const float* wrow = w_qkv + (size_t)(256 + h * 32 + e) * CDIM;
    float acc = 0.0f;
    for (int c = 0; c < CDIM; ++c) acc += srow[c] * wrow[c];
    ctx[idx] = acc;
}

// T[b][o][h*32+d] = sum_e w_out[o][h*32+e] * ctx[b][h][d][e]
__global__ void t_kernel(const float* __restrict__ ctx,
                         const float* __restrict__ w_out,
                         float* __restrict__ T) {
    const int idx = blockIdx.x * 256 + threadIdx.x;   // over 16*16384
    const int b   = idx >> 14;
    const int o   = (idx >> 7) & 127;
    const int i   = idx & 127;
    const int h   = i >> 5;
    const int d   = i & 31;
    const float* wo = w_out + o * CDIM + h * 32;
    const float* cx = ctx + (size_t)b * 4096 + h * 1024 + d * 32;
    float acc = 0.0f;
#pragma unroll
    for (int e = 0; e < 32; ++e) acc += wo[e] * cx[e];
    T[idx] = acc;
}

// Weff[b][o][c] = scale * sum_i T[b][o][i] * Wq[i][c]   (Wq = w_qkv rows 0..127)
__global__ void weff_kernel(const float* __restrict__ T,
                            const float* __restrict__ w_qkv,
                            float* __restrict__ Weff) {
    const int idx = blockIdx.x * 256 + threadIdx.x;   // over 16*16384
    const int b   = idx >> 14;
    const int o   = (idx >> 7) & 127;
    const int c   = idx & 127;
    const float* trow = T + (size_t)b * (CDIM * CDIM) + o * CDIM;
    float acc = 0.0f;
    for (int i = 0; i < CDIM; ++i) acc += trow[i] * w_qkv[i * CDIM + c];
    Weff[idx] = 0.17677669529663687f * acc;   // 32^-0.5
}

// ---------------------------------------------------------------------------
extern "C" void kernel_launch(void* const* d_in, const int* in_sizes, int n_in,
                              void* d_out, int out_size, void* d_ws, size_t ws_size,
                              hipStream_t stream) {
    const float* x      = (const float*)d_in[0];   // (16,128,128,128)
    const float* w_qkv  = (const float*)d_in[1];   // (384,128)
    const float* w_out  = (const float*)d_in[2];   // (128,128)
    const float* b_out  = (const float*)d_in[3];   // (128,)

    float* ws   = (float*)d_ws;
    float* kbuf = ws;                                   // 16*128*16384
    float* rmax = kbuf + (size_t)NBATCH * CDIM * NPIX;  // 2048
    float* rinv = rmax + NBATCH * CDIM;                 // 2048
    float* Spart = rinv + NBATCH * CDIM;                // 16*16*16384
    float* Ssum  = Spart + (size_t)NCHUNKS * NBATCH * CDIM * CDIM; // 16*16384
    float* ctx   = Ssum + (size_t)NBATCH * CDIM * CDIM;            // 16*4096
    float* T     = ctx + (size_t)NBATCH * 4096;                    // 16*16384
    float* Weff  = T + (size_t)NBATCH * CDIM * CDIM;               // 16*16384

    // 1) k = Wk * x  (Wk = w_qkv rows 128..255)
    gemm_wmma_kernel<false><<<dim3(NPIX / 32, NBATCH), 256, 0, stream>>>(
        w_qkv + 128 * CDIM, 0, x, nullptr, kbuf);

    // 2) softmax stats per (b,d) row
    softmax_stats_kernel<<<NBATCH * CDIM, 256, 0, stream>>>(kbuf, rmax, rinv);

    // 3) S = ksoft * x^T  (split-K over n, per-chunk partials)
    s_partial_kernel<<<dim3(4, NCHUNKS, NBATCH), 256, 0, stream>>>(
        kbuf, x, rmax, rinv, Spart);
    s_reduce_kernel<<<(NBATCH * CDIM * CDIM) / 256, 256, 0, stream>>>(Spart, Ssum);

    // 4) tiny per-batch combines: ctx -> T -> W_eff
    ctx_kernel<<<(NBATCH * 4096) / 256, 256, 0, stream>>>(Ssum, w_qkv, ctx);
    t_kernel<<<(NBATCH * CDIM * CDIM) / 256, 256, 0, stream>>>(ctx, w_out, T);
    weff_kernel<<<(NBATCH * CDIM * CDIM) / 256, 256, 0, stream>>>(T, w_qkv, Weff);

    // 5) out = W_eff[b] * x + b_out
    gemm_wmma_kernel<true><<<dim3(NPIX / 32, NBATCH), 256, 0, stream>>>(
        Weff, (size_t)(CDIM * CDIM), x, b_out, (float*)d_out);
}